// ExpertsChooseMaskedRouter_72816875536874
// MI455X (gfx1250) — compile-verified
//
#include <hip/hip_runtime.h>
#include <hip/hip_bf16.h>

typedef float v2f __attribute__((ext_vector_type(2)));
typedef float v4f __attribute__((ext_vector_type(4)));
typedef float v8f __attribute__((ext_vector_type(8)));

constexpr int G = 4, T = 4096, H = 1024, E = 8, CAP = 512;
constexpr size_t OUT_HALF = (size_t)G * T * E * CAP;   // 67,108,864 elements

// ---------------------------------------------------------------------------
// Kernel 1: streaming zero-fill of the two [G,T,E,C] outputs (536 MB total).
// Non-temporal B128 stores: pure store-bandwidth bound (~23 us at 23.3 TB/s),
// and NT keeps the 192MB L2 from being flushed by write-once data.
// ---------------------------------------------------------------------------
__global__ void zero_out_kernel(v4f* __restrict__ out, size_t n4) {
    size_t i      = (size_t)blockIdx.x * blockDim.x + threadIdx.x;
    size_t stride = (size_t)gridDim.x * blockDim.x;
    v4f z = {0.f, 0.f, 0.f, 0.f};
    for (; i < n4; i += stride) {
        __builtin_nontemporal_store(z, out + i);
    }
}

// ---------------------------------------------------------------------------
// Kernel 2: router logits via V_WMMA_F32_16X16X4_F32 + softmax over E=8.
// One wave handles a 16-token tile against a zero-padded 16-wide expert block.
// A-frag (16x4 f32): lanes 0-15 hold K=k,k+1 ; lanes 16-31 hold K=k+2,k+3.
// B-frag (4x16 f32): v0 = row (k+2*half), v1 = row (k+2*half+1), col = lane&15.
// C/D  (16x16 f32): VGPR r, lanes 0-15 -> M=r, N=lane ; lanes 16-31 -> M=r+8.
// Softmax over N is an aligned 8-lane shfl_xor reduction (lanes 0-7 / 16-23).
// Writes probs transposed as [G, E, T] into workspace for the top-k kernel.
// ---------------------------------------------------------------------------
__global__ __launch_bounds__(128) void router_probs_kernel(
    const float* __restrict__ in,     // [G,T,H]
    const float* __restrict__ W,      // [H,E]
    const float* __restrict__ bias,   // [E]
    float* __restrict__ probs_t)      // [G,E,T]
{
    __shared__ float Wlds[H * 16];    // zero-padded weights, 64 KB

    // stage W padded to 16 columns
    for (int i = threadIdx.x; i < H * 16; i += blockDim.x) {
        int col = i & 15;
        Wlds[i] = (col < E) ? W[(i >> 4) * E + col] : 0.0f;
    }
    __syncthreads();

    const int lane  = threadIdx.x & 31;
    const int m     = lane & 15;          // row-in-tile / expert column
    const int half  = lane >> 4;
    const int tile  = blockIdx.x * (blockDim.x >> 5) + (threadIdx.x >> 5);
    const int tok0  = tile * 16;          // global token index [0, G*T)

    const float* arow = in + (size_t)(tok0 + m) * H + 2 * half;
    const float  bval = (m < E) ? bias[m] : 0.0f;

    v8f c = {};
#pragma unroll 4
    for (int k = 0; k < H; k += 4) {
        v2f a, b;
        a.x = arow[k];
        a.y = arow[k + 1];
        const int kr = k + 2 * half;
        b.x = Wlds[kr * 16 + m];
        b.y = Wlds[(kr + 1) * 16 + m];
        c = __builtin_amdgcn_wmma_f32_16x16x4_f32(
                false, a, false, b, (short)0, c, false, false);
    }

#pragma unroll
    for (int r = 0; r < 8; ++r) {
        float x = c[r] + bval;            // logit for token (tok0+r+8*half), expert m
        // max over the 8 valid experts (aligned 8-lane group)
        float mx = x;
        mx = fmaxf(mx, __shfl_xor(mx, 1));
        mx = fmaxf(mx, __shfl_xor(mx, 2));
        mx = fmaxf(mx, __shfl_xor(mx, 4));
        float ex = __expf(x - mx);
        float sm = ex;
        sm += __shfl_xor(sm, 1);
        sm += __shfl_xor(sm, 2);
        sm += __shfl_xor(sm, 4);
        float p = ex / sm;
        if (m < E) {
            int t  = tok0 + r + 8 * half;     // global token
            int g  = t >> 12;                 // / T
            int tl = t & (T - 1);             // % T
            probs_t[((size_t)(g * E + m) << 12) + tl] = p;
        }
    }
}

// ---------------------------------------------------------------------------
// Kernel 3: per-(g,e) top-512-of-4096 via LDS bitonic sort of composite keys,
// then scatter (1.0, gate) into dispatch/combine. Key = (~mono(f32)<<32)|idx:
// ascending sort => descending prob, lowest index wins ties (jax top_k order).
// ---------------------------------------------------------------------------
__global__ __launch_bounds__(1024) void topk_scatter_kernel(
    const float* __restrict__ probs_t,   // [G,E,T]
    float* __restrict__ out)             // dispatch | combine
{
    __shared__ unsigned long long keys[T];   // 32 KB

    const int row = blockIdx.x;          // g*E + e
    const int g   = row >> 3;
    const int e   = row & (E - 1);
    const int tid = threadIdx.x;
    const float* prow = probs_t + ((size_t)row << 12);

    for (int i = tid; i < T; i += blockDim.x) {
        unsigned bits = __float_as_uint(prow[i]);
        unsigned u = (bits & 0x80000000u) ? ~bits : (bits ^ 0x80000000u); // ascending-monotone
        keys[i] = ((unsigned long long)(~u) << 32) | (unsigned)i;
    }
    __syncthreads();

    for (int k = 2; k <= T; k <<= 1) {
        for (int j = k >> 1; j > 0; j >>= 1) {
            for (int i = tid; i < T; i += blockDim.x) {
                int ixj = i ^ j;
                if (ixj > i) {
                    unsigned long long a = keys[i], b = keys[ixj];
                    bool up = ((i & k) == 0);
                    if (up ? (a > b) : (a < b)) { keys[i] = b; keys[ixj] = a; }
                }
            }
            __syncthreads();
        }
    }

    if (tid < CAP) {
        unsigned long long key = keys[tid];
        unsigned idx = (unsigned)(key & 0xFFFFFFFFu);
        unsigned u   = ~(unsigned)(key >> 32);
        unsigned bits = (u & 0x80000000u) ? (u ^ 0x80000000u) : ~u;
        float gate = __uint_as_float(bits);
        size_t off = (((size_t)g * T + idx) * E + e) * CAP + tid;
        out[off]            = 1.0f;   // dispatch_mask
        out[off + OUT_HALF] = gate;   // combine_array
    }
}

// ---------------------------------------------------------------------------
extern "C" void kernel_launch(void* const* d_in, const int* in_sizes, int n_in,
                              void* d_out, int out_size, void* d_ws, size_t ws_size,
                              hipStream_t stream) {
    const float* inputs = (const float*)d_in[0];   // [G,T,H]
    const float* W      = (const float*)d_in[1];   // [H,E]
    const float* bias   = (const float*)d_in[2];   // [E]
    // d_in[3] = expert_capacity (device scalar) -- fixed at 512 per setup.

    float* out     = (float*)d_out;
    float* probs_t = (float*)d_ws;                 // [G,E,T] f32, 512 KB

    // 1) zero both outputs (2 * 67M floats), NT streaming stores
    size_t n4 = ((size_t)out_size) / 4;
    zero_out_kernel<<<4096, 256, 0, stream>>>((v4f*)out, n4);

    // 2) router logits (WMMA f32) + softmax -> probs_t [G,E,T]
    const int tiles = (G * T) / 16;                // 1024 wave-tiles
    router_probs_kernel<<<tiles / 4, 128, 0, stream>>>(inputs, W, bias, probs_t);

    // 3) per-(g,e) top-k sort + scatter of dispatch/combine
    topk_scatter_kernel<<<G * E, 1024, 0, stream>>>(probs_t, out);
}